// RealLoss_40621800686162
// MI455X (gfx1250) — compile-verified
//
#include <hip/hip_runtime.h>

// Matting-Laplacian loss for (B=4, C=3, H=W=512) fp32 inputs.
// Closed-form per-window evaluation (no 9x9 materialization):
//   y^T vals y = Q - (S^2 + t^T inv t)/9,  t = p - S*mu
// LDS-tiled, VALU fp32 for the batched 3x3 algebra (does not fit 16x16 WMMA
// shapes efficiently), V_WMMA_F32_16X16X4_F32 ones-matmul for the wave-level
// reduction, deterministic two-stage final reduce.

namespace {

constexpr int kH = 512;
constexpr int kW = 512;
constexpr int kB = 4;
constexpr int kCWin = kH - 2;                        // 510 windows per dim
constexpr int kTile = 16;
constexpr int kTiles = (kCWin + kTile - 1) / kTile;  // 32 tiles per dim
constexpr int kSRow = kTile + 2;                     // 18 pixels per tile edge
constexpr int kSPad = 20;                            // padded LDS row stride
constexpr float kLambda = 1e-7f / 9.0f;              // EPS / WIN_SIZE

typedef float v2f __attribute__((ext_vector_type(2)));
typedef float v8f __attribute__((ext_vector_type(8)));

__global__ __launch_bounds__(256)
void matting_tile_kernel(const float* __restrict__ xout,  // "output"  (B,3,H,W)
                         const float* __restrict__ img,   // "content" (B,3,H,W)
                         float* __restrict__ partials)    // (B, kTiles*kTiles)
{
  __shared__ float sm[6][kSRow][kSPad];
  __shared__ float wred[8];

  const int tid = threadIdx.x;
  const int b  = blockIdx.z;
  const int x0 = blockIdx.x * kTile;
  const int y0 = blockIdx.y * kTile;

  const float* cbase = img  + (size_t)b * 3 * kH * kW;
  const float* xbase = xout + (size_t)b * 3 * kH * kW;

  // Cooperative load: 6 planes (3 content + 3 output) of an 18x18 pixel tile.
  for (int i = tid; i < 6 * kSRow * kSRow; i += 256) {
    const int p   = i / (kSRow * kSRow);
    const int rem = i - p * (kSRow * kSRow);
    const int r   = rem / kSRow;
    const int c   = rem - r * kSRow;
    const int gy = y0 + r, gx = x0 + c;
    float v = 0.0f;
    if (gy < kH && gx < kW) {
      const float* src = (p < 3) ? (cbase + (size_t)p * kH * kW)
                                 : (xbase + (size_t)(p - 3) * kH * kW);
      v = src[(size_t)gy * kW + gx];
    }
    sm[p][r][c] = v;
  }
  __syncthreads();

  const int ly = tid >> 4;
  const int lx = tid & 15;

  // Sliding-window moments.
  float m0=0.f,m1=0.f,m2=0.f;
  float g00=0.f,g01=0.f,g02=0.f,g11=0.f,g12=0.f,g22=0.f;
  float S0=0.f,S1=0.f,S2=0.f,Q=0.f;
  float p00=0.f,p01=0.f,p02=0.f,p10=0.f,p11=0.f,p12=0.f,p20=0.f,p21=0.f,p22=0.f;

#pragma unroll
  for (int dy = 0; dy < 3; ++dy) {
#pragma unroll
    for (int dx = 0; dx < 3; ++dx) {
      const float c0 = sm[0][ly+dy][lx+dx];
      const float c1 = sm[1][ly+dy][lx+dx];
      const float c2 = sm[2][ly+dy][lx+dx];
      const float u0 = sm[3][ly+dy][lx+dx];
      const float u1 = sm[4][ly+dy][lx+dx];
      const float u2 = sm[5][ly+dy][lx+dx];
      m0 += c0; m1 += c1; m2 += c2;
      g00 = fmaf(c0,c0,g00); g01 = fmaf(c0,c1,g01); g02 = fmaf(c0,c2,g02);
      g11 = fmaf(c1,c1,g11); g12 = fmaf(c1,c2,g12); g22 = fmaf(c2,c2,g22);
      S0 += u0; S1 += u1; S2 += u2;
      Q = fmaf(u0,u0,Q); Q = fmaf(u1,u1,Q); Q = fmaf(u2,u2,Q);
      p00 = fmaf(c0,u0,p00); p01 = fmaf(c0,u1,p01); p02 = fmaf(c0,u2,p02);
      p10 = fmaf(c1,u0,p10); p11 = fmaf(c1,u1,p11); p12 = fmaf(c1,u2,p12);
      p20 = fmaf(c2,u0,p20); p21 = fmaf(c2,u1,p21); p22 = fmaf(c2,u2,p22);
    }
  }

  const float inv9 = 1.0f / 9.0f;
  const float mu0 = m0*inv9, mu1 = m1*inv9, mu2 = m2*inv9;
  const float a00 = g00*inv9 - mu0*mu0 + kLambda;
  const float a01 = g01*inv9 - mu0*mu1;
  const float a02 = g02*inv9 - mu0*mu2;
  const float a11 = g11*inv9 - mu1*mu1 + kLambda;
  const float a12 = g12*inv9 - mu1*mu2;
  const float a22 = g22*inv9 - mu2*mu2 + kLambda;

  // Symmetric 3x3 inverse via adjugate.
  const float d0 = a11*a22 - a12*a12;
  const float d1 = a02*a12 - a01*a22;
  const float d2 = a01*a12 - a02*a11;
  const float det = a00*d0 + a01*d1 + a02*d2;
  const float rdet = 1.0f / det;
  const float i00 = d0*rdet, i01 = d1*rdet, i02 = d2*rdet;
  const float i11 = (a00*a22 - a02*a02)*rdet;
  const float i12 = (a01*a02 - a00*a12)*rdet;
  const float i22 = (a00*a11 - a01*a01)*rdet;

  float contrib = 0.0f;
  {
    const float t0 = p00 - S0*mu0, t1 = p10 - S0*mu1, t2 = p20 - S0*mu2;
    const float q = i00*t0*t0 + i11*t1*t1 + i22*t2*t2
                  + 2.0f*(i01*t0*t1 + i02*t0*t2 + i12*t1*t2);
    contrib += S0*S0 + q;
  }
  {
    const float t0 = p01 - S1*mu0, t1 = p11 - S1*mu1, t2 = p21 - S1*mu2;
    const float q = i00*t0*t0 + i11*t1*t1 + i22*t2*t2
                  + 2.0f*(i01*t0*t1 + i02*t0*t2 + i12*t1*t2);
    contrib += S1*S1 + q;
  }
  {
    const float t0 = p02 - S2*mu0, t1 = p12 - S2*mu1, t2 = p22 - S2*mu2;
    const float q = i00*t0*t0 + i11*t1*t1 + i22*t2*t2
                  + 2.0f*(i01*t0*t1 + i02*t0*t2 + i12*t1*t2);
    contrib += S2*S2 + q;
  }

  float v = Q - contrib * inv9;
  const bool valid = (y0 + ly < kCWin) && (x0 + lx < kCWin);
  v = valid ? v : 0.0f;  // branch-free mask; EXEC stays all-ones for WMMA

  // ---- Wave reduction via V_WMMA_F32_16X16X4_F32 (ones-matmul trick) ----
  // A = all-ones (16x4): every row of C = column sums of B. B's 64 slots are
  // exactly the wave's 2 VGPRs x 32 lanes, so sum(all C columns) = sum(lane
  // values) regardless of the precise B (k,n) lane mapping. C VGPR0 holds two
  // identical rows across lanes 0-15 / 16-31, so a full 32-lane butterfly of
  // csum[0] yields 2x the wave total.
  v2f aones = {1.0f, 1.0f};
  v2f bvals = {v, 0.0f};
  v8f cacc = {};
  v8f csum = __builtin_amdgcn_wmma_f32_16x16x4_f32(
      false, aones, false, bvals, (short)0, cacc, false, false);
  float r = csum[0];
#pragma unroll
  for (int off = 16; off > 0; off >>= 1)
    r += __shfl_xor(r, off, 32);
  const float wave_sum = 0.5f * r;

  if ((tid & 31) == 0) wred[tid >> 5] = wave_sum;
  __syncthreads();
  if (tid == 0) {
    float s = 0.0f;
#pragma unroll
    for (int w = 0; w < 8; ++w) s += wred[w];
    partials[(size_t)b * (kTiles * kTiles) + blockIdx.y * kTiles + blockIdx.x] = s;
  }
}

__global__ __launch_bounds__(256)
void final_reduce_kernel(const float* __restrict__ partials,
                         float* __restrict__ out)
{
  __shared__ double sd[256];
  const int b = blockIdx.x, tid = threadIdx.x;
  const float* p = partials + (size_t)b * (kTiles * kTiles);
  double s = 0.0;
  for (int i = tid; i < kTiles * kTiles; i += 256) s += (double)p[i];
  sd[tid] = s;
  __syncthreads();
  for (int off = 128; off > 0; off >>= 1) {
    if (tid < off) sd[tid] += sd[tid + off];
    __syncthreads();
  }
  if (tid == 0) out[b] = (float)sd[0];
}

} // namespace

extern "C" void kernel_launch(void* const* d_in, const int* in_sizes, int n_in,
                              void* d_out, int out_size, void* d_ws, size_t ws_size,
                              hipStream_t stream) {
  (void)in_sizes; (void)n_in; (void)out_size; (void)ws_size;
  const float* xout    = (const float*)d_in[0];   // "output"
  const float* content = (const float*)d_in[1];   // "content"
  float* partials = (float*)d_ws;                 // 4 * 1024 floats (16 KB)
  float* loss     = (float*)d_out;                // 4 floats (B,1,1)

  dim3 grid1(kTiles, kTiles, kB);
  matting_tile_kernel<<<grid1, 256, 0, stream>>>(xout, content, partials);
  final_reduce_kernel<<<kB, 256, 0, stream>>>(partials, loss);
}